// differentiableSuperpixelTokenizer_71476845740512
// MI455X (gfx1250) — compile-verified
//
#include <hip/hip_runtime.h>
#include <hip/hip_bf16.h>
#include <math.h>

// Problem dimensions (fixed by the reference).
#define NB 8
#define NC 3
#define NH 224
#define NW 224
#define NP (NH * NW)          // 50176 pixels per image
#define NS 196                // segments per image
#define ND 768                // embed dim
#define DCHUNK 64             // channels handled per proj block (LDS = 196*64*4 = 49KB)
#define NCHUNK (ND / DCHUNK)  // 12
#define TILES 4               // pixel tiles per image (P/TILES = 12544 = 98*128)
#define PIX_PER_TILE (NP / TILES)
#define LN_EPS 1e-5f

typedef float v2f __attribute__((ext_vector_type(2)));
typedef float v8f __attribute__((ext_vector_type(8)));

// ---------------------------------------------------------------- zero ws
__global__ void zero_kernel(float* __restrict__ p, int n) {
  int i = blockIdx.x * blockDim.x + threadIdx.x;
  if (i < n) p[i] = 0.0f;
}

// ---------------------------------------------------------------- counts
// One block per 1024 pixels of one image; LDS histogram then global atomics.
__global__ void __launch_bounds__(256)
count_kernel(const int* __restrict__ seg, float* __restrict__ gcount) {
  __shared__ int hist[NS];
  const int b = blockIdx.y;
  const int base = blockIdx.x * 1024;
  for (int i = threadIdx.x; i < NS; i += 256) hist[i] = 0;
  __syncthreads();
  for (int i = threadIdx.x; i < 1024; i += 256) {
    int s = seg[(size_t)b * NP + base + i];
    atomicAdd(&hist[s], 1);
  }
  __syncthreads();
  for (int i = threadIdx.x; i < NS; i += 256) {
    int h = hist[i];
    if (h) atomicAdd(&gcount[b * NS + i], (float)h);
  }
}

// ------------------------------------------------- projection + scatter-sum
// Block = (image b, 64-channel chunk, pixel tile). 8 waves; each wave does
// 16 pixels/iter via V_WMMA_F32_16X16X4_F32 (K: C=3 padded to 4), ReLU,
// then ds_add_f32 into the per-block LDS segment accumulator.
__global__ void __launch_bounds__(256)
proj_scatter(const float* __restrict__ x, const float* __restrict__ w,
             const int* __restrict__ seg, float* __restrict__ gsum) {
  __shared__ float lsum[NS * DCHUNK];  // 49KB

  const int blk = blockIdx.x;
  const int tile = blk % TILES;
  const int chunk = (blk / TILES) % NCHUNK;
  const int b = blk / (TILES * NCHUNK);

  const int tid = threadIdx.x;
  const int lane = tid & 31;
  const int wave = tid >> 5;
  const int n = lane & 15;       // N (col in 16x16 tile) / pixel-m for A loads
  const int khalf = lane >> 4;   // 0 -> holds K=0,1 ; 1 -> holds K=2,3

  for (int i = tid; i < NS * DCHUNK; i += 256) lsum[i] = 0.0f;
  __syncthreads();

  // B fragments: B[k][n] = w_proj[(chunkbase + nc*16 + n)*3 + k], k==3 -> 0.
  // ISA layout (B 4x16): lanes 0-15 K=0(V0),1(V1); lanes 16-31 K=2(V0),3(V1).
  const int chunkbase = chunk * DCHUNK;
  v2f bfrag[4];
#pragma unroll
  for (int nc = 0; nc < 4; ++nc) {
    int ch = chunkbase + nc * 16 + n;
    float b0, b1;
    if (khalf == 0) { b0 = w[ch * 3 + 0]; b1 = w[ch * 3 + 1]; }
    else            { b0 = w[ch * 3 + 2]; b1 = 0.0f; }
    bfrag[nc][0] = b0;
    bfrag[nc][1] = b1;
  }

  const float* xb = x + (size_t)b * NC * NP;
  const int*   sb = seg + (size_t)b * NP;
  const int pix0 = tile * PIX_PER_TILE;

  for (int p = pix0 + wave * 16; p < pix0 + PIX_PER_TILE; p += 8 * 16) {
    const int pix = p + n;  // lanes 0-15 / 16-31 cover the same 16 pixels
    // A fragment (16x4): lane holds X[m=n][k], k = 2*khalf + {0,1}; k==3 -> 0.
    v2f afrag;
    if (khalf == 0) { afrag[0] = xb[0 * NP + pix]; afrag[1] = xb[1 * NP + pix]; }
    else            { afrag[0] = xb[2 * NP + pix]; afrag[1] = 0.0f; }
    const int myseg = sb[pix];  // lane m holds seg of pixel p+m (m = lane&15)

#pragma unroll
    for (int nc = 0; nc < 4; ++nc) {
      v8f acc = {};
      acc = __builtin_amdgcn_wmma_f32_16x16x4_f32(
          /*neg_a=*/false, afrag, /*neg_b=*/false, bfrag[nc],
          /*c_mod=*/(short)0, acc, /*reuse_a=*/false, /*reuse_b=*/false);
      // D layout: lane L, VGPR r -> M = r + 8*(L>=16), N = L&15.
#pragma unroll
      for (int r = 0; r < 8; ++r) {
        int m = r + 8 * khalf;
        int sm = __shfl(myseg, m, 32);  // ds_bpermute
        float v = acc[r];
        v = v > 0.0f ? v : 0.0f;        // ReLU before pooling
        atomicAdd(&lsum[sm * DCHUNK + nc * 16 + n], v);  // ds_add_f32
      }
    }
  }
  __syncthreads();

  // Flush block-local sums into the global accumulator.
  for (int i = tid; i < NS * DCHUNK; i += 256) {
    float v = lsum[i];
    if (v != 0.0f) {
      int s = i / DCHUNK, c = i % DCHUNK;
      atomicAdd(&gsum[((size_t)b * NS + s) * ND + chunkbase + c], v);
    }
  }
}

// -------------------------------------------- mean + pos-embed + LayerNorm
// One block per (b, s) row; 256 threads * 3 channels each.
__global__ void __launch_bounds__(256)
finalize_kernel(const float* __restrict__ gsum, const float* __restrict__ gcount,
                const int* __restrict__ cent, const float* __restrict__ gamma,
                const float* __restrict__ beta, float* __restrict__ out) {
  const int row = blockIdx.x;  // b*S + s
  const int tid = threadIdx.x;
  const float inv = 1.0f / fmaxf(gcount[row], 1.0f);
  const float c0 = (float)cent[row * 2 + 0] / (float)(NH - 1);  // used by d>=384
  const float c1 = (float)cent[row * 2 + 1] / (float)(NW - 1);  // used by d<384
  const float twopi = 6.283185307179586f;

  float vals[3];
  float lsumv = 0.0f, lsq = 0.0f;
#pragma unroll
  for (int j = 0; j < 3; ++j) {
    int d = tid + j * 256;
    float coord = (d < 384) ? c1 : c0;
    int t = ((d < 384) ? d : (d - 384)) >> 1;              // 0..191
    float e = (float)(2 * (t >> 1)) / 384.0f;              // 2*floor(t/2)/num_pos_feats
    float dimt = powf(10000.0f, e);
    float ang = coord * twopi / dimt;
    float pe = (d & 1) ? cosf(ang) : sinf(ang);
    float v = gsum[(size_t)row * ND + d] * inv + pe;
    vals[j] = v;
    lsumv += v;
    lsq += v * v;
  }

  __shared__ float red[32];
  for (int off = 16; off > 0; off >>= 1) {
    lsumv += __shfl_down(lsumv, off, 32);
    lsq   += __shfl_down(lsq, off, 32);
  }
  const int wave = tid >> 5, lane = tid & 31;
  if (lane == 0) { red[wave] = lsumv; red[8 + wave] = lsq; }
  __syncthreads();
  if (wave == 0) {
    float s = (lane < 8) ? red[lane] : 0.0f;
    float q = (lane < 8) ? red[8 + lane] : 0.0f;
    for (int off = 4; off > 0; off >>= 1) {
      s += __shfl_down(s, off, 32);
      q += __shfl_down(q, off, 32);
    }
    if (lane == 0) { red[16] = s; red[17] = q; }
  }
  __syncthreads();
  const float mean = red[16] / (float)ND;
  const float var = red[17] / (float)ND - mean * mean;
  const float rstd = rsqrtf(var + LN_EPS);
#pragma unroll
  for (int j = 0; j < 3; ++j) {
    int d = tid + j * 256;
    out[(size_t)row * ND + d] = (vals[j] - mean) * rstd * gamma[d] + beta[d];
  }
}

// ---------------------------------------------------------------- launch
extern "C" void kernel_launch(void* const* d_in, const int* in_sizes, int n_in,
                              void* d_out, int out_size, void* d_ws, size_t ws_size,
                              hipStream_t stream) {
  (void)in_sizes; (void)n_in; (void)out_size; (void)ws_size;
  const float* features  = (const float*)d_in[0];  // [8,3,224,224]
  const float* w_proj    = (const float*)d_in[1];  // [768,3]
  const float* ln_gamma  = (const float*)d_in[2];  // [768]
  const float* ln_beta   = (const float*)d_in[3];  // [768]
  const int*   segments  = (const int*)d_in[4];    // [8,224,224]
  const int*   centroids = (const int*)d_in[5];    // [8,196,2]
  float* out = (float*)d_out;                      // [8,196,768]

  float* gsum   = (float*)d_ws;                    // B*S*D floats (~4.8 MB)
  float* gcount = gsum + (size_t)NB * NS * ND;     // B*S floats

  const int nz = NB * NS * ND + NB * NS;
  zero_kernel<<<(nz + 255) / 256, 256, 0, stream>>>(gsum, nz);

  count_kernel<<<dim3(NP / 1024, NB), 256, 0, stream>>>(segments, gcount);

  proj_scatter<<<NB * NCHUNK * TILES, 256, 0, stream>>>(features, w_proj,
                                                        segments, gsum);

  finalize_kernel<<<NB * NS, 256, 0, stream>>>(gsum, gcount, centroids,
                                               ln_gamma, ln_beta, out);
}